// AnomalyAttention_75153337745815
// MI455X (gfx1250) — compile-verified
//
#include <hip/hip_runtime.h>
#include <hip/hip_bf16.h>

// ---------------------------------------------------------------------------
// AnomalyAttention (Anomaly Transformer) for MI455X / gfx1250.
//
// Shapes: B=4, L=1024, H=8, E=D=64.
// Key insight: irfftn(s=(H,L,L)) crops the L-freq axis to 8, keeps only the
// 8 H-freqs and 33 E-rfft freqs -> only 8*8*33 complex modes per batch
// survive. Everything reduces to small separable DFT projections plus a
// 33-term cosine synthesis per output element. The problem is write-BW bound
// (411 MB of outputs -> ~25-40us at 23.3 TB/s); compute is negligible.
// The einsum series@values runs on v_wmma_f32_16x16x4_f32 (full f32).
// ---------------------------------------------------------------------------

#define TWO_PI      6.283185307179586f
#define LN3         1.0986122886681098f
#define INV_SQRT2PI 0.3989422804014327f
#define INV_N23     (1.0f / 8388608.0f)   // 1/(8*1024*1024)
#define SM_SCALE    0.125f                // 1/sqrt(64)

enum : int { B_ = 4, L_ = 1024, H_ = 8, E_ = 64, D_ = 64, NF3 = 33 };

typedef float v2f __attribute__((ext_vector_type(2)));
typedef float v8f __attribute__((ext_vector_type(8)));

// ---------------------------------------------------------------------------
// K1: forward DFT projections.
// C[b,f1,f2,f3] = sum_{l,h,e} x[b,l,h,e] e^{-2pi i(f1 l/1024 + f2 h/8 + f3 e/64)}
// grid: (f3=33, b=4, tensor=2); block: 256 threads.
// ---------------------------------------------------------------------------
__global__ __launch_bounds__(256)
void k1_forward_dft(const float* __restrict__ q,
                    const float* __restrict__ k,
                    float* __restrict__ Cq,
                    float* __restrict__ Ck) {
  __shared__ float tcos[1024];
  __shared__ float tsin[1024];
  __shared__ float pre[256][8];
  __shared__ float pim[256][8];

  const int tid = threadIdx.x;
  const int f3  = blockIdx.x;
  const int b   = blockIdx.y;
  const float* src = (blockIdx.z == 0) ? q : k;
  float*       dst = (blockIdx.z == 0) ? Cq : Ck;

  for (int m = tid; m < 1024; m += 256) {
    float ang = (TWO_PI / 1024.0f) * (float)m;
    __sincosf(ang, &tsin[m], &tcos[m]);
  }
  __syncthreads();

  float accR = 0.0f, accI = 0.0f;  // valid for tid < 64: f1 = tid>>3, f2 = tid&7

  for (int chunk = 0; chunk < 4; ++chunk) {
    const int l = chunk * 256 + tid;
    const float* row = src + ((size_t)b * L_ + l) * (H_ * E_);  // 512 contiguous

    // e-DFT per h: a[h] = sum_e x * e^{-2pi i f3 e/64}  (phase idx = 16*f3*e mod 1024)
    float ar[8], ai[8];
#pragma unroll
    for (int h = 0; h < 8; ++h) { ar[h] = 0.0f; ai[h] = 0.0f; }
    for (int h = 0; h < 8; ++h) {
      const float* hr = row + h * 64;
      for (int e = 0; e < 64; ++e) {
        float x = hr[e];
        int m = (f3 * e * 16) & 1023;
        ar[h] += x * tcos[m];
        ai[h] -= x * tsin[m];
      }
    }
    // h-DFT: p[f2] = sum_h a[h] e^{-2pi i f2 h/8}  (phase idx = 128*f2*h mod 1024)
#pragma unroll
    for (int f2 = 0; f2 < 8; ++f2) {
      float pr = 0.0f, pq = 0.0f;
#pragma unroll
      for (int h = 0; h < 8; ++h) {
        int m = (f2 * h * 128) & 1023;
        float c = tcos[m], s = tsin[m];
        pr += ar[h] * c + ai[h] * s;
        pq += ai[h] * c - ar[h] * s;
      }
      pre[tid][f2] = pr;
      pim[tid][f2] = pq;
    }
    __syncthreads();

    // l-DFT partial reduce: 64 threads own (f1,f2) combos
    if (tid < 64) {
      int f1 = tid >> 3, f2 = tid & 7;
      for (int ll = 0; ll < 256; ++ll) {
        int lcur = chunk * 256 + ll;
        int m = (f1 * lcur) & 1023;
        float c = tcos[m], s = tsin[m];
        float xr = pre[ll][f2], xi = pim[ll][f2];
        accR += xr * c + xi * s;
        accI += xi * c - xr * s;
      }
    }
    __syncthreads();
  }

  if (tid < 64) {
    int f1 = tid >> 3, f2 = tid & 7;
    size_t idx = ((((size_t)b * 8 + f1) * 8 + f2) * NF3 + f3) * 2;
    dst[idx + 0] = accR;
    dst[idx + 1] = accI;
  }
}

// ---------------------------------------------------------------------------
// K2: X = Q*conj(K); Z[i,f2,f3] = sum_f1 X e^{+2pi i f1 i/8};
//     W[b,i,j,f3] = sum_f2 Z e^{+2pi i f2 j/1024}.
// grid: (jchunk=4, i=8, b=4); block: 256 threads. W interleaved (re,im).
// ---------------------------------------------------------------------------
__global__ __launch_bounds__(256)
void k2_build_w(const float* __restrict__ Cq,
                const float* __restrict__ Ck,
                float* __restrict__ W) {
  __shared__ float zr[8][NF3];
  __shared__ float zi[8][NF3];

  const int tid = threadIdx.x;
  const int i   = blockIdx.y;
  const int b   = blockIdx.z;

  for (int c = tid; c < 8 * NF3; c += 256) {
    int f2 = c / NF3, f3 = c % NF3;
    float sr = 0.0f, si = 0.0f;
#pragma unroll
    for (int f1 = 0; f1 < 8; ++f1) {
      size_t idx = ((((size_t)b * 8 + f1) * 8 + f2) * NF3 + f3) * 2;
      float qr = Cq[idx], qi = Cq[idx + 1];
      float kr = Ck[idx], ki = Ck[idx + 1];
      float xr = qr * kr + qi * ki;     // Q * conj(K)
      float xi = qi * kr - qr * ki;
      float ang = (TWO_PI / 8.0f) * (float)((f1 * i) & 7);
      float cc, ss;
      __sincosf(ang, &ss, &cc);
      sr += xr * cc - xi * ss;
      si += xr * ss + xi * cc;
    }
    zr[f2][f3] = sr;
    zi[f2][f3] = si;
  }
  __syncthreads();

  const int j = blockIdx.x * 256 + tid;
  float wc, ws;
  __sincosf((TWO_PI / 1024.0f) * (float)j, &ws, &wc);  // w = e^{+2pi i j/1024}

  float* out = W + ((((size_t)b * 8 + i) * L_ + j) * NF3) * 2;
  for (int f3 = 0; f3 < NF3; ++f3) {
    float ur = 1.0f, ui = 0.0f;
    float sr = 0.0f, si = 0.0f;
#pragma unroll
    for (int f2 = 0; f2 < 8; ++f2) {
      sr += zr[f2][f3] * ur - zi[f2][f3] * ui;
      si += zr[f2][f3] * ui + zi[f2][f3] * ur;
      float nr = ur * wc - ui * ws;
      ui = ur * ws + ui * wc;
      ur = nr;
    }
    out[2 * f3 + 0] = sr;
    out[2 * f3 + 1] = si;
  }
}

// ---------------------------------------------------------------------------
// K3: per row (b,h,j): y_k = sum_f3 c_f3 (Wre cos - Wim sin)(2pi f3 k/1024),
// softmax over k, * 1/sqrt(E). grid: 32768 rows; block: 256 threads (4 k each).
// ---------------------------------------------------------------------------
__global__ __launch_bounds__(256)
void k3_series(const float* __restrict__ W,
               float* __restrict__ series) {
  __shared__ float tcos[1024];
  __shared__ float tsin[1024];
  __shared__ float wre[NF3];
  __shared__ float wim[NF3];
  __shared__ float red[256];

  const int tid = threadIdx.x;
  const int row = blockIdx.x;  // ((b*8+h)*1024 + j)

  for (int m = tid; m < 1024; m += 256) {
    float ang = (TWO_PI / 1024.0f) * (float)m;
    __sincosf(ang, &tsin[m], &tcos[m]);
  }
  if (tid < NF3) {
    const float* wr = W + ((size_t)row * NF3) * 2;
    float cf = (tid == 0) ? INV_N23 : (2.0f * INV_N23);  // fold irfft weights + 1/2^23
    wre[tid] = wr[2 * tid + 0] * cf;
    wim[tid] = wr[2 * tid + 1] * cf;
  }
  __syncthreads();

  float y[4];
#pragma unroll
  for (int r = 0; r < 4; ++r) {
    int kk = tid + 256 * r;
    float acc = 0.0f;
    for (int f3 = 0; f3 < NF3; ++f3) {
      int m = (f3 * kk) & 1023;
      acc += wre[f3] * tcos[m] - wim[f3] * tsin[m];
    }
    y[r] = acc;
  }

  // block max
  float mx = fmaxf(fmaxf(y[0], y[1]), fmaxf(y[2], y[3]));
  red[tid] = mx;
  __syncthreads();
  for (int s = 128; s > 0; s >>= 1) {
    if (tid < s) red[tid] = fmaxf(red[tid], red[tid + s]);
    __syncthreads();
  }
  mx = red[0];
  __syncthreads();

  float sum = 0.0f;
#pragma unroll
  for (int r = 0; r < 4; ++r) {
    y[r] = __expf(y[r] - mx);
    sum += y[r];
  }
  red[tid] = sum;
  __syncthreads();
  for (int s = 128; s > 0; s >>= 1) {
    if (tid < s) red[tid] += red[tid + s];
    __syncthreads();
  }
  float inv = SM_SCALE / red[0];

  float* out = series + (size_t)row * L_;
#pragma unroll
  for (int r = 0; r < 4; ++r) out[tid + 256 * r] = y[r] * inv;
}

// ---------------------------------------------------------------------------
// K4: prior + sigma4. grid: 32768 rows (b,h,i); block: 256 threads, float4 j.
// ---------------------------------------------------------------------------
__global__ __launch_bounds__(256)
void k4_prior(const float* __restrict__ sigma,
              float* __restrict__ prior,
              float* __restrict__ sigma4) {
  const int row = blockIdx.x;
  const int i  = row & 1023;
  const int bh = row >> 10;
  const int b  = bh >> 3, h = bh & 7;

  float sg = sigma[((size_t)b * L_ + i) * H_ + h];
  float s  = 1.0f / (1.0f + __expf(-5.0f * sg)) + 1e-5f;
  s = __expf(s * LN3) - 1.0f;                 // 3^s - 1
  float amp = INV_SQRT2PI / s;
  float neg_inv2s2 = -0.5f / (s * s);

  const int j0 = threadIdx.x * 4;
  float4 p;
  {
    float d0 = (float)(i - (j0 + 0)); p.x = amp * __expf(neg_inv2s2 * d0 * d0);
    float d1 = (float)(i - (j0 + 1)); p.y = amp * __expf(neg_inv2s2 * d1 * d1);
    float d2 = (float)(i - (j0 + 2)); p.z = amp * __expf(neg_inv2s2 * d2 * d2);
    float d3 = (float)(i - (j0 + 3)); p.w = amp * __expf(neg_inv2s2 * d3 * d3);
  }
  ((float4*)(prior  + (size_t)row * L_))[threadIdx.x] = p;
  ((float4*)(sigma4 + (size_t)row * L_))[threadIdx.x] = make_float4(s, s, s, s);
}

// ---------------------------------------------------------------------------
// K5: V[b,l,h,d] = sum_s series[b,h,l,s] * values[b,s,h,d]
// via v_wmma_f32_16x16x4_f32. grid: (ltile32=32, bh=32); block: 256 = 8 waves;
// each wave owns one 16x16 C tile (2 l-subtiles x 4 d-tiles per block).
// ---------------------------------------------------------------------------
__global__ __launch_bounds__(256)
void k5_gemm(const float* __restrict__ series,
             const float* __restrict__ values,
             float* __restrict__ V) {
  const int tid  = threadIdx.x;
  const int wave = tid >> 5;
  const int lane = tid & 31;
  const int bh   = blockIdx.y;
  const int b    = bh >> 3, h = bh & 7;
  const int l0   = blockIdx.x * 32 + (wave >> 2) * 16;
  const int d0   = (wave & 3) * 16;

  const float* A  = series + (size_t)bh * L_ * L_;            // [l][s] row-major
  const float* Bv = values + (size_t)b * L_ * (H_ * D_) + h * D_;  // [s][d], stride 512

  const int n    = lane & 15;         // A row in tile / B column in tile
  const int ksel = (lane >> 4) * 2;   // K sub-offset 0 or 2

  v8f acc = {};
  for (int k0 = 0; k0 < L_; k0 += 4) {
    const float* ap = A + (size_t)(l0 + n) * L_ + (k0 + ksel);
    v2f a;
    a.x = ap[0];
    a.y = ap[1];
    const float* bp = Bv + (size_t)(k0 + ksel) * (H_ * D_) + (d0 + n);
    v2f bb;
    bb.x = bp[0];
    bb.y = bp[H_ * D_];
    acc = __builtin_amdgcn_wmma_f32_16x16x4_f32(false, a, false, bb,
                                                (short)0, acc, false, false);
  }

  // C/D layout: VGPR r -> lanes 0-15: M=r, lanes 16-31: M=r+8; N = lane&15
  const int mbase = l0 + 8 * (lane >> 4);
#pragma unroll
  for (int r = 0; r < 8; ++r) {
    int l = mbase + r;
    V[(((size_t)b * L_ + l) * H_ + h) * D_ + d0 + n] = acc[r];
  }
}

// ---------------------------------------------------------------------------
extern "C" void kernel_launch(void* const* d_in, const int* in_sizes, int n_in,
                              void* d_out, int out_size, void* d_ws, size_t ws_size,
                              hipStream_t stream) {
  const float* q  = (const float*)d_in[0];  // (B,L,H,E)
  const float* k  = (const float*)d_in[1];  // (B,L,H,E)
  const float* v  = (const float*)d_in[2];  // (B,L,H,D)
  const float* sg = (const float*)d_in[3];  // (B,L,H)

  float* out    = (float*)d_out;
  float* Vout   = out;                              // B*L*H*D   = 2,097,152
  float* series = Vout + (size_t)B_ * L_ * H_ * D_; // B*H*L*L   = 33,554,432
  float* prior  = series + (size_t)B_ * H_ * L_ * L_;
  float* sigma4 = prior + (size_t)B_ * H_ * L_ * L_;

  float* ws = (float*)d_ws;
  float* Cq = ws;                              // 4*8*8*33*2 = 16,896 floats
  float* Ck = Cq + (size_t)B_ * 8 * 8 * NF3 * 2;
  float* W  = Ck + (size_t)B_ * 8 * 8 * NF3 * 2;  // 4*8*1024*33*2 = 2,162,688 floats

  k1_forward_dft<<<dim3(NF3, B_, 2), 256, 0, stream>>>(q, k, Cq, Ck);
  k2_build_w<<<dim3(4, 8, B_), 256, 0, stream>>>(Cq, Ck, W);
  k3_series<<<B_ * H_ * L_, 256, 0, stream>>>(W, series);
  k4_prior<<<B_ * H_ * L_, 256, 0, stream>>>(sg, prior, sigma4);
  k5_gemm<<<dim3(32, 32), 256, 0, stream>>>(series, v, Vout);
}